// SinusoidalTransformer_71459665871290
// MI455X (gfx1250) — compile-verified
//
#include <hip/hip_runtime.h>
#include <math.h>

// Problem constants (B,C,H,W,E) = (64,3,64,64,128)
#define NB    64
#define NCHW  12288          // 3*64*64
#define NE    128
#define CPW   4              // chw points per workgroup
#define NWG   (NCHW / CPW)   // 3072 workgroups
#define BLK   256            // 4 chw groups x 64 batch lanes

// ---------------- CDNA5 hardware helpers (guarded) ----------------

// v_sin_f32 / v_cos_f32 compute sin/cos of (2*pi*x); we pre-fold 1/(2pi)
// into the coefficients so the inner loop hits the trans unit directly.
__device__ __forceinline__ float hw_sin_2pi(float a) {
#if defined(__AMDGCN__) && __has_builtin(__builtin_amdgcn_sinf)
  return __builtin_amdgcn_sinf(a);
#else
  return sinf(a * 6.28318530717958647692f);
#endif
}
__device__ __forceinline__ float hw_cos_2pi(float a) {
#if defined(__AMDGCN__) && __has_builtin(__builtin_amdgcn_cosf)
  return __builtin_amdgcn_cosf(a);
#else
  return cosf(a * 6.28318530717958647692f);
#endif
}

__device__ __forceinline__ void wait_async_zero() {
#if defined(__AMDGCN__)
#if __has_builtin(__builtin_amdgcn_s_wait_asynccnt)
  __builtin_amdgcn_s_wait_asynccnt(0);
#else
  asm volatile("s_wait_asynccnt 0" ::: "memory");
#endif
#endif
}

#if defined(__AMDGCN__) && __has_builtin(__builtin_amdgcn_global_load_async_to_lds_b128)
#define HAVE_ASYNC_LDS 1
typedef __attribute__((ext_vector_type(4))) int v4i;
// Builtin signature (from clang diagnostic): first param is
// "int __vector(4) __device__ *" i.e. v4i addrspace(1)*; LDS side is
// v4i addrspace(3)*; then immediate offset and cpol.
__device__ __forceinline__ void async_b128(const float* g, float* l) {
  typedef __attribute__((address_space(1))) v4i gv4i_t;
  typedef __attribute__((address_space(3))) v4i lv4i_t;
  __builtin_amdgcn_global_load_async_to_lds_b128((gv4i_t*)g, (lv4i_t*)l, 0, 0);
}
#else
#define HAVE_ASYNC_LDS 0
#endif

// 64-thread (2-wave) reduction inside one chw group; all 256 threads must call.
template <bool IS_MAX>
__device__ __forceinline__ float groupReduce(float v, float* red, int ty, int tb) {
  red[ty * 64 + tb] = v;
  __syncthreads();
  float r;
  if (tb < 32) {  // threads [ty*64, ty*64+32): exactly one full wave32
    float a = red[ty * 64 + tb];
    float b = red[ty * 64 + tb + 32];
    r = IS_MAX ? fmaxf(a, b) : (a + b);
#pragma unroll
    for (int m = 16; m >= 1; m >>= 1) {
      float o = __shfl_xor(r, m, 32);
      r = IS_MAX ? fmaxf(r, o) : (r + o);
    }
    if (tb == 0) red[ty * 64] = r;
  }
  __syncthreads();
  r = red[ty * 64];
  __syncthreads();
  return r;
}

// ---------------- main fused kernel ----------------
// Block layout: t = ty*64 + tb ; ty in [0,4) selects chw, tb in [0,64) = batch.
__global__ __launch_bounds__(BLK) void sinflow_main_kernel(
    const float* __restrict__ x,       // (B, CHW)
    const float* __restrict__ in_w,    // (CHW, E)  raw (pre-softplus)
    const float* __restrict__ in_b,    // (CHW, E)
    const float* __restrict__ out_w,   // (CHW, E)  raw (pre-softmax)
    const float* __restrict__ out_b,   // (CHW,)
    const float* __restrict__ res_w,   // (CHW,)
    float* __restrict__ y,             // (B, CHW)
    float* __restrict__ partial) {     // (NWG, NB) log-det partials
  __shared__ float  s_inw[CPW * NE];
  __shared__ float  s_inb[CPW * NE];
  __shared__ float  s_ow [CPW * NE];
  __shared__ float4 s_p  [CPW * NE];   // {w/pi, b/pi, coef, mix}
  __shared__ float  s_red[CPW * 64];
  __shared__ float  s_c0[CPW], s_res[CPW], s_ob[CPW];

  const int t  = threadIdx.x;
  const int ty = t >> 6;
  const int tb = t & 63;
  const size_t base = (size_t)blockIdx.x * (CPW * NE);

  // ---- Stage raw params global -> LDS (CDNA5 async-to-LDS path) ----
#if HAVE_ASYNC_LDS
  if (t < (CPW * NE) / 4) {  // 128 threads x one b128 per array
    async_b128(in_w  + base + t * 4, &s_inw[t * 4]);
    async_b128(in_b  + base + t * 4, &s_inb[t * 4]);
    async_b128(out_w + base + t * 4, &s_ow [t * 4]);
  }
  wait_async_zero();
#else
  for (int i = t; i < CPW * NE; i += BLK) {
    s_inw[i] = in_w[base + i];
    s_inb[i] = in_b[base + i];
    s_ow [i] = out_w[base + i];
  }
#endif
  __syncthreads();

  // ---- B-invariant preprocessing: each thread owns e = 2*tb, 2*tb+1 ----
  const int chw = blockIdx.x * CPW + ty;
  const int e0  = 2 * tb;
  float iw0 = s_inw[ty * NE + e0], iw1 = s_inw[ty * NE + e0 + 1];
  float ib0 = s_inb[ty * NE + e0], ib1 = s_inb[ty * NE + e0 + 1];
  float ow0 = s_ow [ty * NE + e0], ow1 = s_ow [ty * NE + e0 + 1];

  // softplus (stable): max(x,0) + log1p(exp(-|x|))  -> strictly > 0
  float w0 = fmaxf(iw0, 0.f) + log1pf(__expf(-fabsf(iw0)));
  float w1 = fmaxf(iw1, 0.f) + log1pf(__expf(-fabsf(iw1)));

  // softmax over E
  float mx    = groupReduce<true >(fmaxf(ow0, ow1), s_red, ty, tb);
  float p0    = __expf(ow0 - mx), p1 = __expf(ow1 - mx);
  float denom = groupReduce<false>(p0 + p1, s_red, ty, tb);
  float rden  = 1.f / denom;
  float mix0 = p0 * rden, mix1 = p1 * rden;

  float coef0 = mix0 * (0.5f / w0);
  float coef1 = mix1 * (0.5f / w1);
  // const0 = sum_e mix * sin(2b) / (2w)
  float c0p = coef0 * __sinf(2.f * ib0) + coef1 * __sinf(2.f * ib1);
  float c0  = groupReduce<false>(c0p, s_red, ty, tb);

  // angle = (2u)/(2pi) = (w*x + b)/pi  -> feed v_sin/v_cos directly
  const float INV_PI = 0.31830988618379067154f;
  s_p[ty * NE + e0    ] = make_float4(w0 * INV_PI, ib0 * INV_PI, coef0, mix0);
  s_p[ty * NE + e0 + 1] = make_float4(w1 * INV_PI, ib1 * INV_PI, coef1, mix1);
  if (tb == 0) {
    s_c0 [ty] = c0;
    s_res[ty] = tanhf(res_w[chw]);
    s_ob [ty] = out_b[chw];
  }
  __syncthreads();

  // ---- batch loop: lane tb owns batch b = tb; wave-uniform LDS broadcast ----
  const float xv = x[(size_t)tb * NCHW + chw];
  float acc_s = 0.f, acc_t = 0.f;
  const float4* __restrict__ P = &s_p[ty * NE];
#pragma unroll 8
  for (int e = 0; e < NE; ++e) {
    float4 q  = P[e];                        // ds_load_b128, same addr per wave
    float ang = fmaf(q.x, xv, q.y);          // (2u)/(2pi)
    acc_s = fmaf(q.z, hw_sin_2pi(ang), acc_s);
    acc_t = fmaf(q.w, hw_cos_2pi(ang), acc_t);
  }
  const float res = s_res[ty];
  y[(size_t)tb * NCHW + chw] = xv + res * (s_c0[ty] - acc_s) + s_ob[ty];
  const float ld = log1pf(-res * acc_t);     // |dy/dx| < 1 so 1+dy > 0

  // per-workgroup log-det partial: sum over the 4 chw groups per batch lane
  s_red[ty * 64 + tb] = ld;
  __syncthreads();
  if (ty == 0) {
    partial[(size_t)blockIdx.x * NB + tb] =
        s_red[tb] + s_red[64 + tb] + s_red[128 + tb] + s_red[192 + tb];
  }
}

// ---------------- tiny final reduction: (NWG, NB) -> (NB) ----------------
__global__ __launch_bounds__(BLK) void logdet_reduce_kernel(
    const float* __restrict__ partial, float* __restrict__ log_dets, int nwg) {
  __shared__ float red[BLK];
  const int b = blockIdx.x;
  float s = 0.f;
  for (int g = threadIdx.x; g < nwg; g += blockDim.x)
    s += partial[(size_t)g * NB + b];
  red[threadIdx.x] = s;
  __syncthreads();
  for (int k = BLK / 2; k >= 1; k >>= 1) {
    if (threadIdx.x < k) red[threadIdx.x] += red[threadIdx.x + k];
    __syncthreads();
  }
  if (threadIdx.x == 0) log_dets[b] = red[0];
}

// ---------------- harness entry ----------------
extern "C" void kernel_launch(void* const* d_in, const int* in_sizes, int n_in,
                              void* d_out, int out_size, void* d_ws, size_t ws_size,
                              hipStream_t stream) {
  const float* x    = (const float*)d_in[0];  // (64,3,64,64)
  const float* inw  = (const float*)d_in[1];  // (3,64,64,128)
  const float* inb  = (const float*)d_in[2];  // (3,64,64,128)
  const float* outw = (const float*)d_in[3];  // (3,64,64,128)
  const float* outb = (const float*)d_in[4];  // (3,64,64)
  const float* resw = (const float*)d_in[5];  // (3,64,64)

  float* y  = (float*)d_out;                  // 786432 floats
  float* ld = y + (size_t)NB * NCHW;          // + 64 floats (log_dets)
  float* partial = (float*)d_ws;              // NWG*NB floats = 768 KB

  hipLaunchKernelGGL(sinflow_main_kernel, dim3(NWG), dim3(BLK), 0, stream,
                     x, inw, inb, outw, outb, resw, y, partial);
  hipLaunchKernelGGL(logdet_reduce_kernel, dim3(NB), dim3(BLK), 0, stream,
                     partial, ld, NWG);
}